// CenterLoss_66623532695554
// MI455X (gfx1250) — compile-verified
//
#include <hip/hip_runtime.h>
#include <hip/hip_bf16.h>

// CenterLoss, MI455X (gfx1250, wave32).
// Pipeline:
//  K1: stream feats once; per-(batch-chunk, 32-dim slab) LDS segment-sum via
//      ds_add_f32; flush non-atomic partials to scratch; per-block sum(f^2).
//  K2a: reduce per-chunk class counts.
//  K2b: reduce partial sums across chunks with V_WMMA_F32_16X16X4_F32
//       (A = ones 16x4 -> every output row = column sum), then form
//       new_centers on the fly and accumulate dot & count-weighted norm terms.
//  K3: deterministic final reduction -> loss scalar.

typedef __attribute__((ext_vector_type(2))) float v2f;
typedef __attribute__((ext_vector_type(8))) float v8f;

#define D_DIM 256
#define SLABD 32
#define NSLAB 8
#define LROW  33          // padded LDS row stride (floats) to spread banks
#define K2B_BLOCKS 128

__global__ __launch_bounds__(256) void cl_k1_segsum(
    const float* __restrict__ feats, const int* __restrict__ labels,
    float* __restrict__ pS, float* __restrict__ pCnt, float* __restrict__ pQ,
    int B, int C, int Pb)
{
  extern __shared__ float smem[];
  float* lsum = smem;             // C * LROW accumulators for this 32-dim slab
  float* lcnt = smem + C * LROW;  // C class counts (used by slab 0 only)
  const int tid  = threadIdx.x;
  const int pb   = blockIdx.x;    // batch chunk
  const int slab = blockIdx.y;    // 32-dim column slab

  for (int i = tid; i < C * (LROW + 1); i += 256) smem[i] = 0.0f;
  __syncthreads();

  const int g    = tid >> 3;            // 32 sample-groups in flight
  const int li   = tid & 7;             // 8 lanes x float4 = 32 dims
  const int col0 = slab * SLABD + li * 4;
  const int spb  = B / Pb;
  const int i0   = pb * spb;
  const int iend = min(B, i0 + spb);

  float ssq = 0.0f;
  for (int i = i0 + g; i < iend; i += 32) {
    // speculative prefetch of the stream (global_prefetch_b8); OOB is dropped
    __builtin_prefetch(&feats[(size_t)(i + 256) * D_DIM + col0], 0, 1);
    const int lab = labels[i];
    const float4 v = *(const float4*)(feats + (size_t)i * D_DIM + col0);
    ssq += v.x * v.x + v.y * v.y + v.z * v.z + v.w * v.w;
    float* row = lsum + lab * LROW + li * 4;
    atomicAdd(row + 0, v.x);            // ds_add_f32
    atomicAdd(row + 1, v.y);
    atomicAdd(row + 2, v.z);
    atomicAdd(row + 3, v.w);
    if (slab == 0 && li == 0) atomicAdd(&lcnt[lab], 1.0f);
  }
  __syncthreads();

  // non-atomic flush of this chunk's slab partials: pS[pb][c][slab*32+d]
  for (int idx = tid; idx < C * SLABD; idx += 256) {
    const int c = idx >> 5, d = idx & 31;
    pS[(size_t)pb * ((size_t)C * D_DIM) + (size_t)c * D_DIM + slab * SLABD + d] =
        lsum[c * LROW + d];
  }
  if (slab == 0)
    for (int c = tid; c < C; c += 256) pCnt[(size_t)pb * C + c] = lcnt[c];

  // per-block sum(f^2) partial (fixed-order tree -> deterministic)
  for (int off = 16; off; off >>= 1) ssq += __shfl_xor(ssq, off, 32);
  __shared__ float red[8];
  if ((tid & 31) == 0) red[tid >> 5] = ssq;
  __syncthreads();
  if (tid == 0) {
    float s = 0.0f;
    for (int w = 0; w < 8; ++w) s += red[w];
    pQ[pb * NSLAB + slab] = s;
  }
}

__global__ void cl_k2a_counts(const float* __restrict__ pCnt,
                              float* __restrict__ gcnt, int C, int Pb)
{
  const int c = blockIdx.x * blockDim.x + threadIdx.x;
  if (c < C) {
    float s = 0.0f;
    for (int pb = 0; pb < Pb; ++pb) s += pCnt[(size_t)pb * C + c];
    gcnt[c] = s;
  }
}

__global__ __launch_bounds__(256) void cl_k2b_wmma(
    const float* __restrict__ pS, const float* __restrict__ gcnt,
    const float* __restrict__ centers, float* __restrict__ pR,
    int C, int Pb)
{
  const size_t CD = (size_t)C * D_DIM;
  const int nchunks = (int)(CD >> 4);         // 16 columns per chunk
  const int lane = threadIdx.x & 31;
  const int wid  = (blockIdx.x * blockDim.x + threadIdx.x) >> 5;  // wave id
  const int nw   = (gridDim.x * blockDim.x) >> 5;
  const int j  = lane & 15;                   // output column within tile
  const int kh = lane >> 4;                   // K half select

  v2f a; a[0] = 1.0f; a[1] = 1.0f;            // A = ones 16x4 -> row-sum matmul
  float dacc = 0.0f, cacc = 0.0f;

  for (int t = wid; t < nchunks; t += nw) {   // wave-uniform loop (EXEC all 1s)
    const int col = t * 16 + j;
    v8f acc = {0.f, 0.f, 0.f, 0.f, 0.f, 0.f, 0.f, 0.f};
    for (int pb0 = 0; pb0 < Pb; pb0 += 4) {
      v2f b;                                  // B tile: 4 chunk-partials x 16 cols
      b[0] = pS[(size_t)(pb0 + kh)     * CD + col];
      b[1] = pS[(size_t)(pb0 + 2 + kh) * CD + col];
      acc = __builtin_amdgcn_wmma_f32_16x16x4_f32(
          false, a, false, b, (short)0, acc, false, false);
    }
    const float total = acc[0];               // every D row = column sum over Pb
    if (lane < 16) {                          // lanes 16..31 are duplicates
      const int c = col / D_DIM;
      const float cnt  = gcnt[c];
      const float cen  = centers[col];
      const float mean = total / fmaxf(cnt, 1.0f);
      const float nc   = (cnt > 0.0f) ? 0.5f * cen + 0.5f * mean : cen;
      dacc += total * nc;                     // Σ_c sums_c · newc_c (per element)
      cacc += cnt * nc * nc;                  // Σ_c cnt_c · ||newc_c||²
    }
  }
  for (int off = 16; off; off >>= 1) {
    dacc += __shfl_xor(dacc, off, 32);
    cacc += __shfl_xor(cacc, off, 32);
  }
  __shared__ float redd[8], redc[8];
  if (lane == 0) { redd[threadIdx.x >> 5] = dacc; redc[threadIdx.x >> 5] = cacc; }
  __syncthreads();
  if (threadIdx.x == 0) {
    float sd = 0.0f, sc = 0.0f;
    for (int w = 0; w < 8; ++w) { sd += redd[w]; sc += redc[w]; }
    pR[blockIdx.x * 2 + 0] = sd;
    pR[blockIdx.x * 2 + 1] = sc;
  }
}

__global__ __launch_bounds__(256) void cl_k3_final(
    const float* __restrict__ pQ, int nQ,
    const float* __restrict__ pR, int nRblocks,
    float* __restrict__ out, int B)
{
  const int tid = threadIdx.x;
  float ssq = 0.0f, dot = 0.0f, cn = 0.0f;
  for (int i = tid; i < nQ; i += 256) ssq += pQ[i];
  for (int i = tid; i < nRblocks; i += 256) { dot += pR[2 * i]; cn += pR[2 * i + 1]; }
  for (int off = 16; off; off >>= 1) {
    ssq += __shfl_xor(ssq, off, 32);
    dot += __shfl_xor(dot, off, 32);
    cn  += __shfl_xor(cn,  off, 32);
  }
  __shared__ float r[3][8];
  if ((tid & 31) == 0) { int w = tid >> 5; r[0][w] = ssq; r[1][w] = dot; r[2][w] = cn; }
  __syncthreads();
  if (tid == 0) {
    float A = 0.0f, Bt = 0.0f, Ct = 0.0f;
    for (int w = 0; w < 8; ++w) { A += r[0][w]; Bt += r[1][w]; Ct += r[2][w]; }
    out[0] = 0.5f * (A - 2.0f * Bt + Ct) / (float)B;
  }
}

extern "C" void kernel_launch(void* const* d_in, const int* in_sizes, int n_in,
                              void* d_out, int out_size, void* d_ws, size_t ws_size,
                              hipStream_t stream)
{
  (void)n_in; (void)out_size;
  const float* feats   = (const float*)d_in[0];
  const float* centers = (const float*)d_in[1];
  const int*   labels  = (const int*)d_in[2];
  const int B = in_sizes[0] / D_DIM;
  const int C = in_sizes[1] / D_DIM;
  const size_t CD = (size_t)C * D_DIM;

  // pick largest batch-chunk count whose scratch fits in d_ws (deterministic)
  int Pb = 4;
  const int cands[4] = {32, 16, 8, 4};
  for (int k = 0; k < 4; ++k) {
    const size_t need = sizeof(float) *
        ((size_t)cands[k] * (CD + (size_t)C) + (size_t)C +
         (size_t)cands[k] * NSLAB + 2 * (size_t)K2B_BLOCKS + 64);
    if (need <= ws_size) { Pb = cands[k]; break; }
  }

  float* ws   = (float*)d_ws;
  float* pS   = ws;                          // [Pb][C*D] chunk partial sums
  float* pCnt = pS + (size_t)Pb * CD;        // [Pb][C]   chunk partial counts
  float* gcnt = pCnt + (size_t)Pb * C;       // [C]       total counts
  float* pQ   = gcnt + C;                    // [Pb*NSLAB] sum(f^2) partials
  float* pR   = pQ + (size_t)Pb * NSLAB;     // [K2B_BLOCKS][2] dot/cnorm partials

  const size_t smem = sizeof(float) * (size_t)C * (LROW + 1);  // 136 KB @ C=1000
  (void)hipFuncSetAttribute((const void*)cl_k1_segsum,
                            hipFuncAttributeMaxDynamicSharedMemorySize, (int)smem);

  cl_k1_segsum<<<dim3(Pb, NSLAB), 256, smem, stream>>>(feats, labels, pS, pCnt, pQ, B, C, Pb);
  cl_k2a_counts<<<(C + 255) / 256, 256, 0, stream>>>(pCnt, gcnt, C, Pb);
  cl_k2b_wmma<<<K2B_BLOCKS, 256, 0, stream>>>(pS, gcnt, centers, pR, C, Pb);
  cl_k3_final<<<1, 256, 0, stream>>>(pQ, Pb * NSLAB, pR, K2B_BLOCKS, (float*)d_out, B);
}